// BayesLinear_91087666413947
// MI455X (gfx1250) — compile-verified
//
#include <hip/hip_runtime.h>

#define NS    2048
#define DIN   512
#define DOUT  512

typedef float v2f __attribute__((ext_vector_type(2)));
typedef float v4f __attribute__((ext_vector_type(4)));
typedef float v8f __attribute__((ext_vector_type(8)));

// ---------------------------------------------------------------------------
// Kernel 1: s[i,o] = exp(weight_log_sigma[i,o])   (512x512 -> d_ws, 1 MB)
// ---------------------------------------------------------------------------
__global__ void prep_sigma_kernel(const float* __restrict__ wls,
                                  float* __restrict__ s) {
    int idx = blockIdx.x * blockDim.x + threadIdx.x;
    s[idx] = __expf(wls[idx]);
}

// ---------------------------------------------------------------------------
// Kernel 2: out[n,o] = bias_mu[o] + exp(bias_log_sigma[o]) * noise_b[n,o]
// (initializes d_out with the bias term)
// ---------------------------------------------------------------------------
__global__ void bias_kernel(const float* __restrict__ bmu,
                            const float* __restrict__ bls,
                            const float* __restrict__ nb,
                            float* __restrict__ out) {
    int idx = blockIdx.x * blockDim.x + threadIdx.x;
    int o = idx & (DOUT - 1);
    out[idx] = bmu[o] + __expf(bls[o]) * nb[idx];
}

// ---------------------------------------------------------------------------
// Kernel 3: out += inputs @ weight_mu using V_WMMA_F32_16X16X4_F32 (exact f32).
// One wave computes a 16x64 output strip (4 adjacent 16x16 tiles, A reused).
// Lane layout per ISA 7.12.2:
//   A (16x4):  lane (lo,hi): v0 = A[lo][2*hi],   v1 = A[lo][2*hi+1]
//   B (4x16):  lane (lo,hi): v0 = B[2*hi][lo],   v1 = B[2*hi+1][lo]
//   C (16x16): VGPR r, lanes 0-15 -> row r, lanes 16-31 -> row r+8, col = lo
// ---------------------------------------------------------------------------
__global__ void __launch_bounds__(256)
gemm_mu_wmma_kernel(const float* __restrict__ X,
                    const float* __restrict__ Wmu,
                    float* __restrict__ out) {
    const int wave = (blockIdx.x * blockDim.x + threadIdx.x) >> 5;
    const int lane = threadIdx.x & 31;
    const int lo   = lane & 15;
    const int hi   = lane >> 4;

    const int mt = wave >> 3;        // 0..127  (m tile of 16)
    const int og = wave & 7;         // 0..7    (o group of 64)
    const int m0 = mt * 16;
    const int o0 = og * 64;
    const int m  = m0 + lo;

    v8f c[4];
    // load C from d_out (bias term already resident there)
    #pragma unroll
    for (int t = 0; t < 4; ++t)
        #pragma unroll
        for (int r = 0; r < 8; ++r)
            c[t][r] = out[(size_t)(m0 + r + 8 * hi) * DOUT + o0 + 16 * t + lo];

    for (int kk = 0; kk < DIN; kk += 4) {
        const int ka = kk + 2 * hi;
        v2f a;
        a.x = X[(size_t)m * DIN + ka];
        a.y = X[(size_t)m * DIN + ka + 1];
        #pragma unroll
        for (int t = 0; t < 4; ++t) {
            v2f b;
            b.x = Wmu[(size_t)ka * DOUT + o0 + 16 * t + lo];
            b.y = Wmu[(size_t)(ka + 1) * DOUT + o0 + 16 * t + lo];
            // 8 args: (neg_a, A, neg_b, B, c_mod, C, reuse_a, reuse_b)
            c[t] = __builtin_amdgcn_wmma_f32_16x16x4_f32(
                false, a, false, b, (short)0, c[t], false, false);
        }
    }

    #pragma unroll
    for (int t = 0; t < 4; ++t)
        #pragma unroll
        for (int r = 0; r < 8; ++r)
            out[(size_t)(m0 + r + 8 * hi) * DOUT + o0 + 16 * t + lo] = c[t][r];
}

// ---------------------------------------------------------------------------
// Kernel 4: out[n,o] += sum_i x[n,i] * s[i,o] * noise_w[n,i,o]
// One block per sample n (128 threads, thread t owns outputs 4t..4t+3).
// noise_w streamed once with NT hint (2.15 GB -> HBM-bound, ~92 us floor);
// s (1 MB) stays hot in the 192 MB L2 across all 2048 blocks.
// ---------------------------------------------------------------------------
__global__ void __launch_bounds__(128)
noise_stream_kernel(const float* __restrict__ x,
                    const float* __restrict__ nw,
                    const float* __restrict__ s,
                    float* __restrict__ out) {
    __shared__ float xs[DIN];
    const int n = blockIdx.x;
    const int t = threadIdx.x;                 // 0..127

    // stage x row (2 KB) into LDS
    ((v4f*)xs)[t] = ((const v4f*)(x + (size_t)n * DIN))[t];
    __syncthreads();

    const v4f* nwp = (const v4f*)(nw + (size_t)n * DIN * DOUT) + t;
    const v4f* sp  = (const v4f*)s + t;

    v4f acc = (v4f)(0.0f);
    #pragma unroll 8
    for (int i = 0; i < DIN; ++i) {
        v4f w  = __builtin_nontemporal_load(nwp + (size_t)i * (DOUT / 4));
        v4f sv = sp[(size_t)i * (DOUT / 4)];
        float xi = xs[i];
        acc += (xi * sv) * w;
    }

    v4f* op = (v4f*)(out + (size_t)n * DOUT) + t;
    *op = *op + acc;
}

// ---------------------------------------------------------------------------
extern "C" void kernel_launch(void* const* d_in, const int* in_sizes, int n_in,
                              void* d_out, int out_size, void* d_ws, size_t ws_size,
                              hipStream_t stream) {
    const float* x   = (const float*)d_in[0];  // inputs            (2048,512)
    const float* nw  = (const float*)d_in[1];  // noise_w           (2048,512,512)
    const float* nb  = (const float*)d_in[2];  // noise_b           (2048,512)
    const float* wmu = (const float*)d_in[3];  // weight_mu         (512,512)
    const float* wls = (const float*)d_in[4];  // weight_log_sigma  (512,512)
    const float* bmu = (const float*)d_in[5];  // bias_mu           (512)
    const float* bls = (const float*)d_in[6];  // bias_log_sigma    (512)
    float* out = (float*)d_out;
    float* s   = (float*)d_ws;                 // 512*512 floats = 1 MB scratch

    // 1. s = exp(weight_log_sigma)
    prep_sigma_kernel<<<(DIN * DOUT) / 256, 256, 0, stream>>>(wls, s);
    // 2. out = bias term
    bias_kernel<<<(NS * DOUT) / 256, 256, 0, stream>>>(bmu, bls, nb, out);
    // 3. out += inputs @ weight_mu  (WMMA f32): 1024 waves = 128 blocks x 8 waves
    gemm_mu_wmma_kernel<<<(NS / 16) * (DOUT / 64) / 8, 256, 0, stream>>>(x, wmu, out);
    // 4. out += streamed per-sample noise contraction (HBM-bound)
    noise_stream_kernel<<<NS, 128, 0, stream>>>(x, nw, s, out);
}